// GaussianSequenceLabeling_14156212208435
// MI455X (gfx1250) — compile-verified
//
#include <hip/hip_runtime.h>
#include <math.h>

#define DIM   48
#define DIM2  96
#define NBATCH 8
#define TLEN  64
#define NLAB  32

typedef __attribute__((ext_vector_type(2))) float v2f;
typedef __attribute__((ext_vector_type(8))) float v8f;

__device__ __forceinline__ v8f wmma_f32_16x16x4(v2f a, v2f b, v8f c) {
  // V_WMMA_F32_16X16X4_F32 : D = A(16x4) * B(4x16) + C(16x16), wave32
  return __builtin_amdgcn_wmma_f32_16x16x4_f32(
      false, a, false, b, (short)0, c, false, false);
}

// ---------------------------------------------------------------------------
// Kernel 1: lam0 = inv(trans_cho^T trans_cho) = G*G^T with G = inv(trans_cho)
//           (trans_cho is lower triangular), eta0 = lam0 @ trans_mu.
// ---------------------------------------------------------------------------
__global__ __launch_bounds__(256) void precompute_kernel(
    const float* __restrict__ tcho, const float* __restrict__ tmu,
    float* __restrict__ lam0, float* __restrict__ eta0)
{
  __shared__ float G[DIM2*DIM2];   // 36 KB
  __shared__ float tmpv[DIM2];
  int tid  = threadIdx.x;
  int lane = tid & 31;
  int wv   = tid >> 5;

  for (int p = tid; p < DIM2*DIM2; p += 256) G[p] = 0.f;
  __syncthreads();

  // Triangular inverse: thread j computes column j of G = T^-1.
  if (tid < DIM2) {
    int j = tid;
    G[j*DIM2 + j] = 1.0f / tcho[j*DIM2 + j];
    for (int i = j + 1; i < DIM2; ++i) {
      float s = 0.f;
      for (int k = j; k < i; ++k) s += tcho[i*DIM2 + k] * G[k*DIM2 + j];
      G[i*DIM2 + j] = -s / tcho[i*DIM2 + i];
    }
  }
  __syncthreads();

  // lam0 = G * G^T via WMMA f32 16x16x4 (6x6 tiles, K=96)
  for (int tile = wv; tile < 36; tile += 8) {
    int tm = tile / 6, tn = tile % 6;
    v8f c = {};
    int m = tm*16 + (lane & 15);
    int n = tn*16 + (lane & 15);
    for (int k0 = 0; k0 < DIM2; k0 += 4) {
      int kk = k0 + ((lane >> 4) << 1);
      v2f av, bv;
      av.x = G[m*DIM2 + kk];      av.y = G[m*DIM2 + kk + 1];
      bv.x = G[n*DIM2 + kk];      bv.y = G[n*DIM2 + kk + 1];   // B = G^T
      c = wmma_f32_16x16x4(av, bv, c);
    }
    int col    = tn*16 + (lane & 15);
    int rowoff = tm*16 + ((lane >> 4) << 3);
    for (int v = 0; v < 8; ++v)
      lam0[(rowoff + v)*DIM2 + col] = c[v];
  }

  // eta0 = G * (G^T * mu) using only LDS-resident G
  if (tid < DIM2) {
    int k = tid; float s = 0.f;
    for (int i = k; i < DIM2; ++i) s += G[i*DIM2 + k] * tmu[i];
    tmpv[k] = s;
  }
  __syncthreads();
  if (tid < DIM2) {
    int i = tid; float s = 0.f;
    for (int k = 0; k <= i; ++k) s += G[i*DIM2 + k] * tmpv[k];
    eta0[i] = s;
  }
}

// ---------------------------------------------------------------------------
// Kernel 2: forward/backward scans in information form.
// One workgroup per (direction, batch). Per step:
//   A = E + Lam ; chol(A) ; X = A^-1 * Bc ; y = A^-1 * (etaE0 + eta)
//   Lam' = K - Bc^T X + diag(1/cho^2)   (GEMM via WMMA)
//   eta' = etaK0 - Bc^T y + mu/cho^2
// Carry at the *start* of each step is stored at slot [dir][b][word].
// ---------------------------------------------------------------------------
__global__ __launch_bounds__(256) void scan_kernel(
    const int*   __restrict__ sentences, const int* __restrict__ border,
    const float* __restrict__ em_mu,     const float* __restrict__ em_cho,
    const float* __restrict__ lam0,      const float* __restrict__ eta0,
    float* __restrict__ lamS, float* __restrict__ etaS)
{
  __shared__ float E [DIM*DIM];
  __shared__ float Bc[DIM*DIM];   // lam0[elim, keep]
  __shared__ float Kb[DIM*DIM];
  __shared__ float A [DIM*DIM];
  __shared__ float X [DIM*52];    // padded stride
  __shared__ float Lm[DIM*DIM];   // carry precision
  __shared__ float et[DIM];       // carry eta
  __shared__ float eE0[DIM], eK0[DIM], yv[DIM];

  int tid  = threadIdx.x;
  int lane = tid & 31;
  int wv   = tid >> 5;
  int b    = blockIdx.x & 7;
  int dir  = blockIdx.x >> 3;
  int eoff = dir ? DIM : 0;   // eliminated block
  int koff = dir ? 0 : DIM;   // kept block

  for (int p = tid; p < DIM*DIM; p += 256) {
    int i = p / DIM, j = p % DIM;
    E [p] = lam0[(eoff+i)*DIM2 + (eoff+j)];
    Bc[p] = lam0[(eoff+i)*DIM2 + (koff+j)];
    Kb[p] = lam0[(koff+i)*DIM2 + (koff+j)];
    Lm[p] = (i == j) ? 1.f : 0.f;
  }
  if (tid < DIM) {
    eE0[tid] = eta0[eoff + tid];
    eK0[tid] = eta0[koff + tid];
    et[tid]  = 0.f;
  }
  __syncthreads();

  for (int t = 0; t < TLEN; ++t) {
    int w = dir ? border[b*TLEN + t] : t;
    float* lo = lamS + (((size_t)dir*NBATCH + b)*TLEN + w)*(size_t)(DIM*DIM);
    float* eo = etaS + (((size_t)dir*NBATCH + b)*TLEN + w)*(size_t)DIM;
    for (int p = tid; p < DIM*DIM; p += 256) lo[p] = Lm[p];
    if (tid < DIM) eo[tid] = et[tid];

    for (int p = tid; p < DIM*DIM; p += 256) A[p] = E[p] + Lm[p];
    __syncthreads();

    // Cholesky A = L L^T (in-place lower)
    for (int k = 0; k < DIM; ++k) {
      if (tid == 0) A[k*DIM + k] = sqrtf(A[k*DIM + k]);
      __syncthreads();
      float lkk = A[k*DIM + k];
      for (int i = k + 1 + tid; i < DIM; i += 256) A[i*DIM + k] /= lkk;
      __syncthreads();
      int rem = DIM - 1 - k;
      for (int p = tid; p < rem*rem; p += 256) {
        int j = k + 1 + p / rem;
        int i = k + 1 + p % rem;
        if (i >= j) A[i*DIM + j] -= A[i*DIM + k] * A[j*DIM + k];
      }
      __syncthreads();
    }

    // Triangular solves: 48 RHS columns (one per thread) + eta RHS
    if (tid < DIM) {
      int c = tid;
      for (int i = 0; i < DIM; ++i) {
        float s = Bc[i*DIM + c];
        for (int j = 0; j < i; ++j) s -= A[i*DIM + j] * X[j*52 + c];
        X[i*52 + c] = s / A[i*DIM + i];
      }
      for (int i = DIM - 1; i >= 0; --i) {
        float s = X[i*52 + c];
        for (int j = i + 1; j < DIM; ++j) s -= A[j*DIM + i] * X[j*52 + c];
        X[i*52 + c] = s / A[i*DIM + i];
      }
    } else if (tid == DIM) {
      for (int i = 0; i < DIM; ++i) {
        float s = eE0[i] + et[i];
        for (int j = 0; j < i; ++j) s -= A[i*DIM + j] * yv[j];
        yv[i] = s / A[i*DIM + i];
      }
      for (int i = DIM - 1; i >= 0; --i) {
        float s = yv[i];
        for (int j = i + 1; j < DIM; ++j) s -= A[j*DIM + i] * yv[j];
        yv[i] = s / A[i*DIM + i];
      }
    }
    __syncthreads();

    int widx = sentences[b*TLEN + w];

    // Lam' = Kb - (Bc^T) * X + diag(1/cho^2)  via WMMA (3x3 tiles, K=48)
    for (int tile = wv; tile < 9; tile += 8) {
      int tm = tile / 3, tn = tile % 3;
      v8f c = {};
      int m = tm*16 + (lane & 15);
      int n = tn*16 + (lane & 15);
      for (int k0 = 0; k0 < DIM; k0 += 4) {
        int kk = k0 + ((lane >> 4) << 1);
        v2f av, bv;
        av.x = Bc[kk*DIM + m];       av.y = Bc[(kk+1)*DIM + m];   // A = Bc^T
        bv.x = X [kk*52 + n];        bv.y = X [(kk+1)*52 + n];
        c = wmma_f32_16x16x4(av, bv, c);
      }
      int col    = tn*16 + (lane & 15);
      int rowoff = tm*16 + ((lane >> 4) << 3);
      for (int v = 0; v < 8; ++v) {
        int r = rowoff + v;
        float val = Kb[r*DIM + col] - c[v];
        if (r == col) {
          float ch = em_cho[(size_t)widx*DIM + r];
          val += 1.f / (ch*ch);
        }
        Lm[r*DIM + col] = val;
      }
    }
    // eta' (matvec, cheap)
    if (tid < DIM) {
      int r = tid;
      float s = eK0[r];
      for (int k = 0; k < DIM; ++k) s -= Bc[k*DIM + r] * yv[k];
      float ch = em_cho[(size_t)widx*DIM + r];
      float mu = em_mu [(size_t)widx*DIM + r];
      et[r] = s + mu / (ch*ch);
    }
    __syncthreads();
  }
}

// ---------------------------------------------------------------------------
// Kernel 3: combine + zeta score. One workgroup per (b,t), 128 threads.
//   ec_lam = lam_f + lam_b + diag(word) ; invert (Gauss-Jordan) ; mu0
//   per label: S = var0 + diag(dec_cho^2) ; Cholesky -> logdet + maha
// ---------------------------------------------------------------------------
__global__ __launch_bounds__(128) void score_kernel(
    const int*   __restrict__ sentences,
    const float* __restrict__ em_mu,  const float* __restrict__ em_cho,
    const float* __restrict__ dec_mu, const float* __restrict__ dec_cho,
    const float* __restrict__ lamS,   const float* __restrict__ etaS,
    float* __restrict__ out)
{
  __shared__ float W[DIM*DIM2];     // [48][96] augmented [ec_lam | I] -> [.. | inv]
  __shared__ float ee[DIM];
  __shared__ float mu0[DIM];
  __shared__ float fr[DIM];
  __shared__ float pv;
  __shared__ float S[4][DIM*49];
  __shared__ float df[4][DIM];
  __shared__ float yw[4][DIM];

  int tid  = threadIdx.x;
  int lane = tid & 31;
  int wv   = tid >> 5;
  int b    = blockIdx.x >> 6;
  int t    = blockIdx.x & 63;
  int widx = sentences[b*TLEN + t];

  const float* lf = lamS + (((size_t)0*NBATCH + b)*TLEN + t)*(size_t)(DIM*DIM);
  const float* lb = lamS + (((size_t)1*NBATCH + b)*TLEN + t)*(size_t)(DIM*DIM);
  const float* ef = etaS + (((size_t)0*NBATCH + b)*TLEN + t)*(size_t)DIM;
  const float* eb = etaS + (((size_t)1*NBATCH + b)*TLEN + t)*(size_t)DIM;

  for (int p = tid; p < DIM*DIM; p += 128) {
    int i = p / DIM, j = p % DIM;
    float v = lf[p] + lb[p];
    if (i == j) {
      float ch = em_cho[(size_t)widx*DIM + i];
      v += 1.f / (ch*ch);
    }
    W[i*DIM2 + j]       = v;
    W[i*DIM2 + DIM + j] = (i == j) ? 1.f : 0.f;
  }
  if (tid < DIM) {
    float ch = em_cho[(size_t)widx*DIM + tid];
    float mu = em_mu [(size_t)widx*DIM + tid];
    ee[tid] = ef[tid] + eb[tid] + mu / (ch*ch);
  }
  __syncthreads();

  // Gauss-Jordan (SPD, no pivoting)
  for (int k = 0; k < DIM; ++k) {
    if (tid == 0) pv = 1.f / W[k*DIM2 + k];
    __syncthreads();
    for (int j = tid; j < DIM2; j += 128) W[k*DIM2 + j] *= pv;
    __syncthreads();
    if (tid < DIM) fr[tid] = (tid == k) ? 0.f : W[tid*DIM2 + k];
    __syncthreads();
    for (int p = tid; p < DIM*DIM2; p += 128) {
      int i = p / DIM2, j = p % DIM2;
      if (i != k) W[p] -= fr[i] * W[k*DIM2 + j];
    }
    __syncthreads();
  }
  if (tid < DIM) {
    float s = 0.f;
    for (int j = 0; j < DIM; ++j) s += W[tid*DIM2 + DIM + j] * ee[j];
    mu0[tid] = s;
  }
  __syncthreads();

  // 32 labels: wave wv handles labels wv*8 .. wv*8+7 (uniform loops)
  for (int q = 0; q < 8; ++q) {
    int l = wv*8 + q;
    float* Sw = &S[wv][0];
    for (int p = lane; p < DIM*DIM; p += 32) {
      int i = p / DIM, j = p % DIM;
      float v = W[i*DIM2 + DIM + j];
      if (i == j) { float dc = dec_cho[l*DIM + i]; v += dc*dc; }
      Sw[i*49 + j] = v;
    }
    for (int i = lane; i < DIM; i += 32)
      df[wv][i] = mu0[i] - dec_mu[l*DIM + i];
    __syncthreads();

    for (int k = 0; k < DIM; ++k) {
      if (lane == 0) Sw[k*49 + k] = sqrtf(Sw[k*49 + k]);
      __syncthreads();
      float lkk = Sw[k*49 + k];
      for (int i = k + 1 + lane; i < DIM; i += 32) Sw[i*49 + k] /= lkk;
      __syncthreads();
      int rem = DIM - 1 - k;
      for (int p = lane; p < rem*rem; p += 32) {
        int j = k + 1 + p / rem;
        int i = k + 1 + p % rem;
        if (i >= j) Sw[i*49 + j] -= Sw[i*49 + k] * Sw[j*49 + k];
      }
      __syncthreads();
    }

    if (lane == 0) {
      float ld = 0.f, maha = 0.f;
      for (int i = 0; i < DIM; ++i) {
        ld += logf(Sw[i*49 + i]);
        float s = df[wv][i];
        for (int j = 0; j < i; ++j) s -= Sw[i*49 + j] * yw[wv][j];
        float y = s / Sw[i*49 + i];
        yw[wv][i] = y;
        maha += y * y;
      }
      ld *= 2.f;
      out[((size_t)b*TLEN + t)*NLAB + l] =
          -0.5f * ((float)DIM * 1.8378770664093453f + ld + maha);
    }
    __syncthreads();
  }
}

// ---------------------------------------------------------------------------
extern "C" void kernel_launch(void* const* d_in, const int* in_sizes, int n_in,
                              void* d_out, int out_size, void* d_ws, size_t ws_size,
                              hipStream_t stream) {
  (void)in_sizes; (void)n_in; (void)out_size; (void)ws_size;
  const int*   sentences = (const int*)  d_in[0];
  const int*   border    = (const int*)  d_in[1];
  const float* em_mu     = (const float*)d_in[2];
  const float* em_cho    = (const float*)d_in[3];
  const float* tmu       = (const float*)d_in[4];
  const float* tcho      = (const float*)d_in[5];
  const float* dec_mu    = (const float*)d_in[6];
  const float* dec_cho   = (const float*)d_in[7];
  float* out = (float*)d_out;
  float* ws  = (float*)d_ws;

  float* lam0 = ws;                         // 96*96
  float* eta0 = ws + 9216;                  // 96
  float* lamS = ws + 9216 + 96 + 32;        // 2*8*64*48*48
  float* etaS = lamS + (size_t)2*NBATCH*TLEN*DIM*DIM;  // 2*8*64*48

  precompute_kernel<<<1, 256, 0, stream>>>(tcho, tmu, lam0, eta0);
  scan_kernel<<<16, 256, 0, stream>>>(sentences, border, em_mu, em_cho,
                                      lam0, eta0, lamS, etaS);
  score_kernel<<<NBATCH*TLEN, 128, 0, stream>>>(sentences, em_mu, em_cho,
                                                dec_mu, dec_cho, lamS, etaS, out);
}